// init_6828998001281
// MI455X (gfx1250) — compile-verified
//
#include <hip/hip_runtime.h>

typedef __attribute__((ext_vector_type(2))) float v2f;
typedef __attribute__((ext_vector_type(8))) float v8f;

#define N_EDGES_C      500000
#define HIDDEN_C       128
#define NRAD_C         6
#define EPB_C          128     // edges per block (8 M-tiles of 16)
#define EMB_ROWS_C     100

__device__ __forceinline__ float silu_f(float v) {
    return v * (1.0f / (1.0f + __expf(-v)));
}

__global__ __launch_bounds__(256, 1)
void edge_mlp_wmma_kernel(const int*   __restrict__ x,
                          const float* __restrict__ rbf,
                          const int*   __restrict__ idx_i,
                          const int*   __restrict__ idx_j,
                          const float* __restrict__ emb_weight,
                          const float* __restrict__ W_rbf0,
                          const float* __restrict__ b_rbf0,
                          const float* __restrict__ W_lin,
                          const float* __restrict__ b_lin,
                          const float* __restrict__ W_rbf1,
                          float* __restrict__ out_e1,
                          float* __restrict__ out_e2)
{
    // LDS staging: emb table (51.2KB) + rbf0 panel (64KB) + small stuff  ~121KB
    __shared__ __align__(16) float s_emb [EMB_ROWS_C * HIDDEN_C];
    __shared__ __align__(16) float s_rbf0[EPB_C * HIDDEN_C];
    __shared__ __align__(16) float s_rbf [EPB_C * NRAD_C];
    __shared__ __align__(16) float s_w0  [NRAD_C * HIDDEN_C];
    __shared__ __align__(16) float s_b0  [HIDDEN_C];
    __shared__ int s_xi[EPB_C];
    __shared__ int s_xj[EPB_C];

    const int tid = threadIdx.x;
    const long long e_base = (long long)blockIdx.x * EPB_C;

    // ---- stage phase -------------------------------------------------------
    for (int p = tid; p < EMB_ROWS_C * HIDDEN_C; p += 256) s_emb[p] = emb_weight[p];
    for (int p = tid; p < NRAD_C * HIDDEN_C;    p += 256) s_w0[p]  = W_rbf0[p];
    if (tid < HIDDEN_C) s_b0[tid] = b_rbf0[tid];
    if (tid < EPB_C) {
        long long e = e_base + tid;
        if (e >= N_EDGES_C) e = N_EDGES_C - 1;       // clamp (stores are guarded)
        s_xi[tid] = x[idx_i[e]];
        s_xj[tid] = x[idx_j[e]];
        #pragma unroll
        for (int r = 0; r < NRAD_C; ++r)
            s_rbf[tid * NRAD_C + r] = rbf[e * NRAD_C + r];
    }
    __syncthreads();

    // ---- rbf0 = silu(rbf @ W_rbf0 + b) panel, K=6 so VALU is right --------
    for (int p = tid; p < EPB_C * HIDDEN_C; p += 256) {
        int m = p >> 7;
        int n = p & (HIDDEN_C - 1);
        float a = s_b0[n];
        #pragma unroll
        for (int r = 0; r < NRAD_C; ++r)
            a += s_rbf[m * NRAD_C + r] * s_w0[r * HIDDEN_C + n];
        s_rbf0[p] = silu_f(a);
    }
    __syncthreads();

    // ---- main GEMM: [128 edges, 384] x [384, 128] via f32 WMMA 16x16x4 ----
    const int wave = tid >> 5;          // 8 waves -> 8 N-tiles of 16 columns
    const int lane = tid & 31;
    const int half = lane >> 4;         // selects K pair (A/B) / M group (C)
    const int nloc = lane & 15;
    const int n    = wave * 16 + nloc;  // this lane's output column

    // per-lane LDS row offsets for the A fragments of each M-tile
    int offI[8], offJ[8], offR[8];
    #pragma unroll
    for (int mt = 0; mt < 8; ++mt) {
        int el = mt * 16 + nloc;        // A row within tile = lane&15
        offI[mt] = s_xi[el] * HIDDEN_C;
        offJ[mt] = s_xj[el] * HIDDEN_C;
        offR[mt] = el * HIDDEN_C;
    }

    v8f acc[8];
    #pragma unroll
    for (int mt = 0; mt < 8; ++mt) acc[mt] = (v8f)0.0f;

    const int co = 2 * half;            // column-pair select within K-step

    // K region 0: h[i]  (cat_emb cols 0..127)
    #pragma unroll 2
    for (int kk = 0; kk < 32; ++kk) {
        const int krow = kk * 4 + co;
        v2f b;
        b.x = W_lin[(krow    ) * HIDDEN_C + n];
        b.y = W_lin[(krow + 1) * HIDDEN_C + n];
        #pragma unroll
        for (int mt = 0; mt < 8; ++mt) {
            v2f a = *(const v2f*)&s_emb[offI[mt] + kk * 4 + co];
            acc[mt] = __builtin_amdgcn_wmma_f32_16x16x4_f32(
                false, a, false, b, (short)0, acc[mt], false, false);
        }
    }
    // K region 1: h[j]  (cat_emb cols 128..255)
    #pragma unroll 2
    for (int kk = 0; kk < 32; ++kk) {
        const int krow = 128 + kk * 4 + co;
        v2f b;
        b.x = W_lin[(krow    ) * HIDDEN_C + n];
        b.y = W_lin[(krow + 1) * HIDDEN_C + n];
        #pragma unroll
        for (int mt = 0; mt < 8; ++mt) {
            v2f a = *(const v2f*)&s_emb[offJ[mt] + kk * 4 + co];
            acc[mt] = __builtin_amdgcn_wmma_f32_16x16x4_f32(
                false, a, false, b, (short)0, acc[mt], false, false);
        }
    }
    // K region 2: rbf0 panel (cat_emb cols 256..383)
    #pragma unroll 2
    for (int kk = 0; kk < 32; ++kk) {
        const int krow = 256 + kk * 4 + co;
        v2f b;
        b.x = W_lin[(krow    ) * HIDDEN_C + n];
        b.y = W_lin[(krow + 1) * HIDDEN_C + n];
        #pragma unroll
        for (int mt = 0; mt < 8; ++mt) {
            v2f a = *(const v2f*)&s_rbf0[offR[mt] + kk * 4 + co];
            acc[mt] = __builtin_amdgcn_wmma_f32_16x16x4_f32(
                false, a, false, b, (short)0, acc[mt], false, false);
        }
    }

    // ---- epilogue: bias + silu -> e1 ; fused (rbf @ W_rbf1) gate -> e2 ----
    const float bias = b_lin[n];
    float w1[NRAD_C];
    #pragma unroll
    for (int r = 0; r < NRAD_C; ++r) w1[r] = W_rbf1[r * HIDDEN_C + n];

    #pragma unroll
    for (int mt = 0; mt < 8; ++mt) {
        #pragma unroll
        for (int c = 0; c < 8; ++c) {
            const int el = mt * 16 + half * 8 + c;   // C/D row layout
            const long long e = e_base + el;
            if (e < N_EDGES_C) {
                float e1v = silu_f(acc[mt][c] + bias);
                float g = 0.0f;
                #pragma unroll
                for (int r = 0; r < NRAD_C; ++r)
                    g += s_rbf[el * NRAD_C + r] * w1[r];
                const long long o = e * HIDDEN_C + n;
                out_e1[o] = e1v;
                out_e2[o] = g * e1v;
            }
        }
    }
}

extern "C" void kernel_launch(void* const* d_in, const int* in_sizes, int n_in,
                              void* d_out, int out_size, void* d_ws, size_t ws_size,
                              hipStream_t stream) {
    const int*   x    = (const int*)  d_in[0];
    const float* rbf  = (const float*)d_in[1];
    const int*   ii   = (const int*)  d_in[2];
    const int*   jj   = (const int*)  d_in[3];
    const float* emb  = (const float*)d_in[4];
    const float* W0   = (const float*)d_in[5];
    const float* b0   = (const float*)d_in[6];
    const float* Wl   = (const float*)d_in[7];
    const float* bl   = (const float*)d_in[8];
    const float* W1   = (const float*)d_in[9];

    float* out_e1 = (float*)d_out;
    float* out_e2 = out_e1 + (size_t)N_EDGES_C * HIDDEN_C;

    const int nblocks = (N_EDGES_C + EPB_C - 1) / EPB_C;   // 3907
    edge_mlp_wmma_kernel<<<nblocks, 256, 0, stream>>>(
        x, rbf, ii, jj, emb, W0, b0, Wl, bl, W1, out_e1, out_e2);
}